// GraphEnergyBlock_59476707115409
// MI455X (gfx1250) — compile-verified
//
#include <hip/hip_runtime.h>

// GraphEnergyBlock fwd+bwd for MI455X (gfx1250), fp32 WMMA (V_WMMA_F32_16X16X4_F32).
// Shapes: B=4, N=1024, D=1024, H=16, QK=64, HID=4096. BETA = 1/8.

typedef float v2f __attribute__((ext_vector_type(2)));
typedef float v8f __attribute__((ext_vector_type(8)));

#define BETA_F 0.125f

#define WMMA_F32(a, b, c) \
  __builtin_amdgcn_wmma_f32_16x16x4_f32(false, (a), false, (b), (short)0, (c), false, false)

static __device__ inline v8f v8zero() {
  v8f z;
#pragma unroll
  for (int i = 0; i < 8; ++i) z[i] = 0.0f;
  return z;
}

// A-fragment (16x4 fp32, row-major source): lane&15 = row, lane>>4 selects K-half.
// Lanes 0-15: {K=k, k+1}; lanes 16-31: {K=k+2, k+3}   (ISA 7.12.2 32-bit A layout)
static __device__ inline v2f frag_a(const float* __restrict__ P, int ld, int r0, int k, int lane) {
  const float* p = P + (size_t)(r0 + (lane & 15)) * ld + (k + 2 * (lane >> 4));
  v2f v; v.x = p[0]; v.y = p[1]; return v;
}

// B-fragment (4x16 fp32) from a row-major [K x N] matrix ("NN"):
// lane&15 = column n, lanes 0-15 hold K rows {k,k+1}, lanes 16-31 {k+2,k+3}.
static __device__ inline v2f frag_b_nn(const float* __restrict__ P, int ld, int k, int n0, int lane) {
  const float* p = P + (size_t)(k + 2 * (lane >> 4)) * ld + (n0 + (lane & 15));
  v2f v; v.x = p[0]; v.y = p[ld]; return v;
}

// B-fragment from a row-major [N x K] matrix accessed transposed ("BT"): same
// element mapping as frag_a with rows = n. Vectorized float2 along K.
static __device__ inline v2f frag_b_bt(const float* __restrict__ P, int ld, int n0, int k, int lane) {
  return frag_a(P, ld, n0, k, lane);
}

// ---------------------------------------------------------------------------
// Generic WMMA GEMM: each wave computes a 32x64 tile of C = op(A*B [+ A2*B2]).
// Per K-step: 2 A-frags + 4 B-frags (6 x b64 loads) feed 8 WMMAs -> ~10.9
// FLOP per L2 byte (vs 6.4 for a 16x64 tile); B-frags reused across row-halves.
// BT: 1 -> B stored [N x K] (use B^T), 0 -> B stored [K x N].
// EPI: 0 -> C = scale*acc; 1 -> C = relu(acc), energy += -0.5*sum(relu^2);
//      2 -> C = C - acc (accumulate-subtract into existing output).
// ---------------------------------------------------------------------------
template <int BT, int EPI, int DUAL>
__global__ __launch_bounds__(256) void gemm_wmma(
    const float* __restrict__ A, const float* __restrict__ B,
    const float* __restrict__ A2, const float* __restrict__ B2,
    float* __restrict__ C, int M, int N, int K,
    int lda, int ldb, int ldc, float scale, float* __restrict__ energy) {
  const int lane = threadIdx.x & 31;
  const int wave = threadIdx.x >> 5;
  const long tile = (long)blockIdx.x * (blockDim.x >> 5) + wave;
  const int tiles_n = N >> 6;
  const int tm = (int)(tile / tiles_n);
  const int tn = (int)(tile % tiles_n);
  if (tm >= (M >> 5)) return;  // wave-uniform guard (EXEC stays all-ones)
  const int m0 = tm << 5;      // 32 rows per wave
  const int n0 = tn << 6;      // 64 cols per wave

  v8f acc[2][4];
#pragma unroll
  for (int r = 0; r < 2; ++r)
#pragma unroll
    for (int j = 0; j < 4; ++j) acc[r][j] = v8zero();

  for (int k = 0; k < K; k += 4) {
    v2f a0 = frag_a(A, lda, m0, k, lane);
    v2f a1 = frag_a(A, lda, m0 + 16, k, lane);
#pragma unroll
    for (int j = 0; j < 4; ++j) {
      v2f b = BT ? frag_b_bt(B, ldb, n0 + 16 * j, k, lane)
                 : frag_b_nn(B, ldb, k, n0 + 16 * j, lane);
      acc[0][j] = WMMA_F32(a0, b, acc[0][j]);
      acc[1][j] = WMMA_F32(a1, b, acc[1][j]);
    }
  }
  if (DUAL) {
    for (int k = 0; k < K; k += 4) {
      v2f a0 = frag_a(A2, lda, m0, k, lane);
      v2f a1 = frag_a(A2, lda, m0 + 16, k, lane);
#pragma unroll
      for (int j = 0; j < 4; ++j) {
        v2f b = BT ? frag_b_bt(B2, ldb, n0 + 16 * j, k, lane)
                   : frag_b_nn(B2, ldb, k, n0 + 16 * j, lane);
        acc[0][j] = WMMA_F32(a0, b, acc[0][j]);
        acc[1][j] = WMMA_F32(a1, b, acc[1][j]);
      }
    }
  }

  // C/D layout: lane&15 = column, VGPR i -> row (8*(lane>>4) + i).
  const int cg = lane >> 4;
  const int cn = lane & 15;
  if (EPI == 1) {
    float s = 0.0f;
#pragma unroll
    for (int r = 0; r < 2; ++r)
#pragma unroll
      for (int j = 0; j < 4; ++j) {
#pragma unroll
        for (int i = 0; i < 8; ++i) {
          float h = acc[r][j][i];
          h = h > 0.0f ? h : 0.0f;
          C[(size_t)(m0 + 16 * r + 8 * cg + i) * ldc + (n0 + 16 * j + cn)] = h;
          s += h * h;
        }
      }
    s += __shfl_xor(s, 1);
    s += __shfl_xor(s, 2);
    s += __shfl_xor(s, 4);
    s += __shfl_xor(s, 8);
    s += __shfl_xor(s, 16);
    if (lane == 0) atomicAdd(energy, -0.5f * s);
  } else if (EPI == 2) {
#pragma unroll
    for (int r = 0; r < 2; ++r)
#pragma unroll
      for (int j = 0; j < 4; ++j)
#pragma unroll
        for (int i = 0; i < 8; ++i) {
          size_t idx = (size_t)(m0 + 16 * r + 8 * cg + i) * ldc + (n0 + 16 * j + cn);
          C[idx] = C[idx] - acc[r][j][i];
        }
  } else {
#pragma unroll
    for (int r = 0; r < 2; ++r)
#pragma unroll
      for (int j = 0; j < 4; ++j)
#pragma unroll
        for (int i = 0; i < 8; ++i)
          C[(size_t)(m0 + 16 * r + 8 * cg + i) * ldc + (n0 + 16 * j + cn)] =
              scale * acc[r][j][i];
  }
}

// ---------------------------------------------------------------------------
// Attention pass 1: per (b,h), each wave owns a 16-row n-tile of queries.
// Streams over key tiles with online softmax (flash style), accumulates
// dq = P*k (unnormalized, rescaled), then writes dq/rowsum, L = max+log(sum),
// and atomically adds e_attn partial = -(1/BETA) * sum(L).
// ---------------------------------------------------------------------------
__global__ __launch_bounds__(256) void attn_dq(
    const float* __restrict__ qbuf, const float* __restrict__ kbuf,
    float* __restrict__ dqbuf, float* __restrict__ Lbuf, float* __restrict__ energy) {
  __shared__ float pts[8][16 * 18];  // per-wave transposed P tile (stride 18: conflict-free, 8B aligned)
  const int lane = threadIdx.x & 31;
  const int wave = threadIdx.x >> 5;
  float* pt = pts[wave];
  const int gw = blockIdx.x * 8 + wave;  // 4096 waves total
  const int bh = gw >> 6;
  const int ntile = gw & 63;
  const int b = bh >> 4, h = bh & 15;
  const float* qh = qbuf + (size_t)b * 1024 * 1024 + h * 64;  // row stride 1024
  const float* kh = kbuf + (size_t)b * 1024 * 1024 + h * 64;
  const int n0 = ntile << 4;

  // Hoist q B-fragments (16 K-steps over QK=64), pre-scaled by BETA.
  v2f qf[16];
#pragma unroll
  for (int t = 0; t < 16; ++t) {
    v2f v = frag_b_bt(qh, 1024, n0, 4 * t, lane);
    v.x *= BETA_F; v.y *= BETA_F;
    qf[t] = v;
  }

  float run_max = -3.0e38f, run_sum = 0.0f;
  v8f dq[4];
#pragma unroll
  for (int j = 0; j < 4; ++j) dq[j] = v8zero();

  for (int mt = 0; mt < 64; ++mt) {
    const int m0 = mt << 4;
    // Score tile D0[m][n] = BETA * k[m].q[n]  (rows m, cols n)
    v8f s = v8zero();
#pragma unroll
    for (int t = 0; t < 16; ++t) {
      v2f a = frag_a(kh, 1024, m0, 4 * t, lane);
      s = WMMA_F32(a, qf[t], s);
    }
    // Column-wise (per n) online softmax: reduce over 8 in-lane rows + half swap.
    float mloc = s[0];
#pragma unroll
    for (int i = 1; i < 8; ++i) mloc = fmaxf(mloc, s[i]);
    mloc = fmaxf(mloc, __shfl_xor(mloc, 16));
    const float nm = fmaxf(run_max, mloc);
    const float corr = __expf(run_max - nm);
    float p[8];
    float ps = 0.0f;
#pragma unroll
    for (int i = 0; i < 8; ++i) { p[i] = __expf(s[i] - nm); ps += p[i]; }
    ps += __shfl_xor(ps, 16);
    run_sum = run_sum * corr + ps;
    run_max = nm;

    // Transpose P through LDS: pt[n][m], n = lane&15, m = 8*(lane>>4)+i.
    float* mp = pt + (lane & 15) * 18 + 8 * (lane >> 4);
#pragma unroll
    for (int i = 0; i < 8; ++i) mp[i] = p[i];
    asm volatile("s_wait_dscnt 0" ::: "memory");

#pragma unroll
    for (int j = 0; j < 4; ++j)
#pragma unroll
      for (int i = 0; i < 8; ++i) dq[j][i] *= corr;

    // dq[n][qk] += sum_m P[n][m] * k[m][qk]
#pragma unroll
    for (int mk = 0; mk < 16; mk += 4) {
      const float* pp = pt + (lane & 15) * 18 + mk + 2 * (lane >> 4);
      v2f pa; pa.x = pp[0]; pa.y = pp[1];
#pragma unroll
      for (int j = 0; j < 4; ++j) {
        v2f kb = frag_b_nn(kh, 1024, m0 + mk, 16 * j, lane);
        dq[j] = WMMA_F32(pa, kb, dq[j]);
      }
    }
    asm volatile("s_wait_dscnt 0" ::: "memory");
  }

  const float Ln = run_max + __logf(run_sum);
  if (lane < 16) Lbuf[(size_t)bh * 1024 + n0 + lane] = Ln;
  float e = Ln;
  e += __shfl_xor(e, 1);
  e += __shfl_xor(e, 2);
  e += __shfl_xor(e, 4);
  e += __shfl_xor(e, 8);
  if (lane == 0) atomicAdd(energy, -8.0f * e);  // -(1/BETA) * sum(L)

  const int cg = lane >> 4;
  const int cn = lane & 15;
#pragma unroll
  for (int j = 0; j < 4; ++j)
#pragma unroll
    for (int i = 0; i < 8; ++i) {
      float rs = __shfl(run_sum, 8 * cg + i, 32);
      dqbuf[(size_t)(b * 1024 + n0 + 8 * cg + i) * 1024 + h * 64 + 16 * j + cn] =
          dq[j][i] / rs;
    }
}

// ---------------------------------------------------------------------------
// Attention pass 2: per (b,h), each wave owns a 16-row m-tile of keys.
// Replays scores transposed using stored L: dk[m] = sum_n exp(s - L[n]) * q[n].
// ---------------------------------------------------------------------------
__global__ __launch_bounds__(256) void attn_dk(
    const float* __restrict__ qbuf, const float* __restrict__ kbuf,
    float* __restrict__ dkbuf, const float* __restrict__ Lbuf) {
  __shared__ float pts[8][16 * 18];
  const int lane = threadIdx.x & 31;
  const int wave = threadIdx.x >> 5;
  float* pt = pts[wave];
  const int gw = blockIdx.x * 8 + wave;
  const int bh = gw >> 6;
  const int mtile = gw & 63;
  const int b = bh >> 4, h = bh & 15;
  const float* qh = qbuf + (size_t)b * 1024 * 1024 + h * 64;
  const float* kh = kbuf + (size_t)b * 1024 * 1024 + h * 64;
  const int m0 = mtile << 4;

  v2f kf[16];
#pragma unroll
  for (int t = 0; t < 16; ++t) {
    v2f v = frag_b_bt(kh, 1024, m0, 4 * t, lane);
    v.x *= BETA_F; v.y *= BETA_F;
    kf[t] = v;
  }

  v8f dk[4];
#pragma unroll
  for (int j = 0; j < 4; ++j) dk[j] = v8zero();

  const int cg = lane >> 4;
  const int cn = lane & 15;

  for (int nt = 0; nt < 64; ++nt) {
    const int n0 = nt << 4;
    // Tile T[n][m] = BETA * q[n].k[m]  (rows n, cols m)
    v8f s = v8zero();
#pragma unroll
    for (int t = 0; t < 16; ++t) {
      v2f a = frag_a(qh, 1024, n0, 4 * t, lane);
      s = WMMA_F32(a, kf[t], s);
    }
    // P = exp(T - L[n]) with L per row.
    float p[8];
#pragma unroll
    for (int i = 0; i < 8; ++i) {
      float Li = Lbuf[(size_t)bh * 1024 + n0 + 8 * cg + i];
      p[i] = __expf(s[i] - Li);
    }
    // Transpose through LDS: pt[m][n], m = cn (col), n = 8*cg+i (row).
    float* mp = pt + cn * 18 + 8 * cg;
#pragma unroll
    for (int i = 0; i < 8; ++i) mp[i] = p[i];
    asm volatile("s_wait_dscnt 0" ::: "memory");

    // dk[m][qk] += sum_n Pt[m][n] * q[n][qk]
#pragma unroll
    for (int nk = 0; nk < 16; nk += 4) {
      const float* pp = pt + cn * 18 + nk + 2 * cg;
      v2f pa; pa.x = pp[0]; pa.y = pp[1];
#pragma unroll
      for (int j = 0; j < 4; ++j) {
        v2f qb = frag_b_nn(qh, 1024, n0 + nk, 16 * j, lane);
        dk[j] = WMMA_F32(pa, qb, dk[j]);
      }
    }
    asm volatile("s_wait_dscnt 0" ::: "memory");
  }

#pragma unroll
  for (int j = 0; j < 4; ++j)
#pragma unroll
    for (int i = 0; i < 8; ++i)
      dkbuf[(size_t)(b * 1024 + m0 + 8 * cg + i) * 1024 + h * 64 + 16 * j + cn] = dk[j][i];
}

__global__ void zero_energy(float* __restrict__ e) {
  if (threadIdx.x == 0 && blockIdx.x == 0) *e = 0.0f;
}

extern "C" void kernel_launch(void* const* d_in, const int* in_sizes, int n_in,
                              void* d_out, int out_size, void* d_ws, size_t ws_size,
                              hipStream_t stream) {
  (void)in_sizes; (void)n_in; (void)ws_size;
  const float* x    = (const float*)d_in[0];  // [4,1024,1024]
  const float* Wq   = (const float*)d_in[1];  // [16,64,1024] -> [1024,1024]
  const float* Wk   = (const float*)d_in[2];  // [16,64,1024] -> [1024,1024]
  const float* Whop = (const float*)d_in[3];  // [1024,4096]

  float* grad   = (float*)d_out;              // [4,1024,1024]
  float* energy = grad + (out_size - 1);      // scalar slot

  float* ws    = (float*)d_ws;
  float* qbuf  = ws;                 // 4096*1024
  float* kbuf  = qbuf + 4194304;     // 4096*1024
  float* dqbuf = kbuf + 4194304;     // 4096*1024
  float* dkbuf = dqbuf + 4194304;    // 4096*1024
  float* Lbuf  = dkbuf + 4194304;    // 64*1024
  float* hbuf  = Lbuf + 65536;       // 4096*1024 (one HID chunk)
  // total workspace: ~80.25 MB

  const int M = 4096;   // B*N
  const int D = 1024;
  const int HID = 4096;

  // 32x64 tiles: (M/32)*(N/64) waves = 2048 -> 256 blocks of 8 waves.
  const int GB = 256;

  zero_energy<<<1, 32, 0, stream>>>(energy);

  // q = x*Wq^T, k = x*Wk^T  ([4096x1024] = [4096x1024]*[1024x1024]^T)
  gemm_wmma<1, 0, 0><<<GB, 256, 0, stream>>>(x, Wq, nullptr, nullptr, qbuf,
                                             M, 1024, D, D, D, 1024, 1.0f, nullptr);
  gemm_wmma<1, 0, 0><<<GB, 256, 0, stream>>>(x, Wk, nullptr, nullptr, kbuf,
                                             M, 1024, D, D, D, 1024, 1.0f, nullptr);

  // Attention energy + dq (P*k), then dk (P^T*q) using stored logsumexp L.
  attn_dq<<<512, 256, 0, stream>>>(qbuf, kbuf, dqbuf, Lbuf, energy);
  attn_dk<<<512, 256, 0, stream>>>(qbuf, kbuf, dkbuf, Lbuf);

  // grad = -(dq @ Wq + dk @ Wk)   (Wq/Wk as [H*QK, D] row-major, NN)
  gemm_wmma<0, 0, 1><<<GB, 256, 0, stream>>>(dqbuf, Wq, dkbuf, Wk, grad,
                                             M, D, 1024, 1024, D, D, -1.0f, nullptr);

  // Hopfield: process HID in 4 chunks of 1024 to bound workspace.
  for (int c = 0; c < 4; ++c) {
    const float* Wc = Whop + c * 1024;
    // h_c = relu(x @ W_hop[:, chunk]), energy += -0.5*sum(h_c^2)
    gemm_wmma<0, 1, 0><<<GB, 256, 0, stream>>>(x, Wc, nullptr, nullptr, hbuf,
                                               M, 1024, D, D, HID, 1024, 1.0f, energy);
    // grad -= h_c @ W_hop[:, chunk]^T
    gemm_wmma<1, 2, 0><<<GB, 256, 0, stream>>>(hbuf, Wc, nullptr, nullptr, grad,
                                               M, D, 1024, 1024, HID, D, 1.0f, nullptr);
  }
}